// RNN_88484916232603
// MI455X (gfx1250) — compile-verified
//
#include <hip/hip_runtime.h>
#include <math.h>

// Problem dimensions (from the reference)
#define HID     128
#define NLAYERS 3
#define NFC     400
#define NL1     500
#define NL2     63
#define NSTEPS  65536
#define NTHREADS 512   // 16 wave32 waves, one WGP

typedef __attribute__((ext_vector_type(2))) float v2f;
typedef __attribute__((ext_vector_type(8))) float v8f;

__device__ __forceinline__ float sigmoid_f(float x) {
    return 1.0f / (1.0f + expf(-x));
}

// Matvec vout[0..nrows) = W[nrows x K] * vin[K] + b0 (+ b1), using
// V_WMMA_F32_16X16X4_F32. One wave handles a 16-row tile; the input vector
// occupies column 0 of the B matrix, so D[:,0] (lanes 0 and 16, acc[0..7])
// holds the 16 dot products. EXEC stays all-1s around every WMMA (all
// conditionals inside the K-loop are v_cndmask selects; tile loop bounds are
// wave-uniform).
//
// skip_fgate: LSTM gate remap — compute only i,g,o rows (W rows >= HID get
// +HID so f-gate rows 128..255 of W/b are never touched); output is compact
// [i(128) | g(128) | o(128)].
__device__ __forceinline__ void mv_wmma(const float* __restrict__ W,
                                        const float* __restrict__ b0,
                                        const float* __restrict__ b1,
                                        const float* __restrict__ vin,
                                        float* __restrict__ vout,
                                        int nrows, int K, bool skip_fgate)
{
    const int lane = threadIdx.x & 31;
    const int wave = threadIdx.x >> 5;
    const int nwav = NTHREADS / 32;
    const int rit  = lane & 15;            // A-matrix row within tile (M)
    const int kof  = (lane >> 4) << 1;     // lanes 0-15 -> K{0,1}; 16-31 -> K{2,3}
    const float bm = (rit == 0) ? 1.0f : 0.0f;  // only lanes 0/16 carry B col 0
    const int ntiles = (nrows + 15) >> 4;

    for (int tile = wave; tile < ntiles; tile += nwav) {
        const int row = tile * 16 + rit;
        const bool rv = row < nrows;
        int wrow = row;
        if (skip_fgate && wrow >= HID) wrow += HID;
        const float* Wr = W + (size_t)(rv ? wrow : 0) * (size_t)K;

        v8f acc = {0.0f, 0.0f, 0.0f, 0.0f, 0.0f, 0.0f, 0.0f, 0.0f};
        for (int k0 = 0; k0 < K; k0 += 4) {
            v2f a, b;
            float a0 = Wr[k0 + kof];
            float a1 = Wr[k0 + kof + 1];
            a.x = rv ? a0 : 0.0f;
            a.y = rv ? a1 : 0.0f;
            // B (4x16, 32-bit): VGPR0 = K{0 | 2}, VGPR1 = K{1 | 3}; col 0 only.
            b.x = vin[k0 + kof]     * bm;
            b.y = vin[k0 + kof + 1] * bm;
            acc = __builtin_amdgcn_wmma_f32_16x16x4_f32(
                      false, a, false, b, (short)0, acc, false, false);
        }

        // D column 0: lane 0 -> rows tile*16+0..7, lane 16 -> rows tile*16+8..15
        if (rit == 0) {
            const int mbase = tile * 16 + ((lane >> 4) << 3);
#pragma unroll
            for (int r = 0; r < 8; ++r) {
                const int orow = mbase + r;
                if (orow < nrows) {
                    int brow = orow;
                    if (skip_fgate && brow >= HID) brow += HID;
                    float v = acc[r] + b0[brow];
                    if (b1) v += b1[brow];
                    vout[orow] = v;
                }
            }
        }
    }
}

__global__ __launch_bounds__(NTHREADS)
void rnn_tail_kernel(const float* __restrict__ inputs,
                     const float* __restrict__ W_ih,
                     const float* __restrict__ b_ih,
                     const float* __restrict__ b_hh,
                     const float* __restrict__ W_fc,
                     const float* __restrict__ b_fc,
                     const float* __restrict__ W_l1,
                     const float* __restrict__ b_l1,
                     const float* __restrict__ W_l2,
                     const float* __restrict__ b_l2,
                     float* __restrict__ out)
{
    __shared__ float vecA[512];
    __shared__ float vecB[512];
    const int tid = threadIdx.x;

    // Warm L2 with the later (larger) weight matrices while the LSTM stages
    // run: ~1.1 MB, one cacheline per prefetch, fire-and-forget.
    {
        const char* pf[3] = { (const char*)W_fc, (const char*)W_l1, (const char*)W_l2 };
        const size_t bytes[3] = { (size_t)NFC * HID * 4,
                                  (size_t)NL1 * NFC * 4,
                                  (size_t)NL2 * NL1 * 4 };
#pragma unroll
        for (int m = 0; m < 3; ++m) {
            for (size_t off = (size_t)tid * 128; off < bytes[m];
                 off += (size_t)NTHREADS * 128)
                __builtin_prefetch(pf[m] + off, 0, 1);
        }
    }

    // Only the last input row matters (state resets every step; output[-1:]).
    if (tid < HID)
        vecA[tid] = inputs[(size_t)(NSTEPS - 1) * HID + tid];
    __syncthreads();

    // 3 LSTM layers, h0=c0=0 => gates = W_ih@x + b_ih + b_hh, f-gate unused.
    for (int l = 0; l < NLAYERS; ++l) {
        const float* Wl = W_ih + (size_t)l * 4 * HID * HID;
        const float* bi = b_ih + (size_t)l * 4 * HID;
        const float* bh = b_hh + (size_t)l * 4 * HID;
        mv_wmma(Wl, bi, bh, vecA, vecB, 3 * HID, HID, true);
        __syncthreads();
        if (tid < HID) {
            const float ig = vecB[tid];
            const float gg = vecB[HID + tid];
            const float og = vecB[2 * HID + tid];
            const float c  = sigmoid_f(ig) * tanhf(gg);
            vecA[tid] = sigmoid_f(og) * tanhf(c);
        }
        __syncthreads();
    }

    // Linear head: FC(400x128) -> L1(500x400) -> L2(63x500), no activations.
    mv_wmma(W_fc, b_fc, nullptr, vecA, vecB, NFC, HID, false);
    __syncthreads();
    mv_wmma(W_l1, b_l1, nullptr, vecB, vecA, NL1, NFC, false);
    __syncthreads();
    mv_wmma(W_l2, b_l2, nullptr, vecA, out, NL2, NL1, false);
}

extern "C" void kernel_launch(void* const* d_in, const int* in_sizes, int n_in,
                              void* d_out, int out_size, void* d_ws, size_t ws_size,
                              hipStream_t stream)
{
    (void)in_sizes; (void)n_in; (void)out_size; (void)d_ws; (void)ws_size;
    const float* inputs = (const float*)d_in[0];
    const float* W_ih   = (const float*)d_in[1];
    // d_in[2] = W_hh: unused (h0 == 0 every step)
    const float* b_ih   = (const float*)d_in[3];
    const float* b_hh   = (const float*)d_in[4];
    const float* W_fc   = (const float*)d_in[5];
    const float* b_fc   = (const float*)d_in[6];
    const float* W_l1   = (const float*)d_in[7];
    const float* b_l1   = (const float*)d_in[8];
    const float* W_l2   = (const float*)d_in[9];
    const float* b_l2   = (const float*)d_in[10];
    float* out = (float*)d_out;

    rnn_tail_kernel<<<1, NTHREADS, 0, stream>>>(
        inputs, W_ih, b_ih, b_hh, W_fc, b_fc, W_l1, b_l1, W_l2, b_l2, out);
}